// NodeAttentionLZeroNormV1_32959579030137
// MI455X (gfx1250) — compile-verified
//
#include <hip/hip_runtime.h>
#include <math.h>

typedef __attribute__((ext_vector_type(2))) float v2f;
typedef __attribute__((ext_vector_type(8))) float v8f;

constexpr int Nn = 32;      // batch
constexpr int C  = 1024;    // channels
constexpr int S  = 1568;    // t*h*w = 32*7*7
constexpr int K  = 128;     // nodes
constexpr int NS = Nn * S;  // 50176 samples per channel
constexpr float EPS = 1e-5f;

__device__ __forceinline__ float block_reduce(float v, float* sm) {
    int tid = threadIdx.x;
    sm[tid] = v;
    __syncthreads();
    for (int off = blockDim.x >> 1; off > 0; off >>= 1) {
        if (tid < off) sm[tid] += sm[tid + off];
        __syncthreads();
    }
    float r = sm[0];
    __syncthreads();
    return r;
}

// ---------------- Stage 1: bn_1 per-channel scale/shift from batch stats ----------------
__global__ __launch_bounds__(256) void k_bn1(const float* __restrict__ x,
                                             const float* __restrict__ g1,
                                             const float* __restrict__ b1,
                                             float* __restrict__ sc1,
                                             float* __restrict__ sh1) {
    __shared__ float sm[256];
    const int c = blockIdx.x, tid = threadIdx.x;
    float s = 0.f, s2 = 0.f;
    for (int n = 0; n < Nn; ++n) {
        const float* p = x + (size_t)(n * C + c) * S;
        for (int i = tid; i < S; i += 256) {
            float v = p[i];
            s += v;
            s2 = fmaf(v, v, s2);
        }
    }
    s  = block_reduce(s,  sm);
    s2 = block_reduce(s2, sm);
    if (tid == 0) {
        float mean = s / (float)NS;
        float var  = s2 / (float)NS - mean * mean;   // biased, torch-style
        float rstd = rsqrtf(var + EPS);
        float sc = g1[c] * rstd;
        sc1[c] = sc;
        sh1[c] = b1[c] - mean * sc;
    }
}

// ---------------- Stage 2a: bn_2 on nodes (stats across K nodes, per channel) ----------------
__global__ __launch_bounds__(256) void k_bn2(const float* __restrict__ nodes,
                                             const float* __restrict__ g2,
                                             const float* __restrict__ b2,
                                             float* __restrict__ ndtmp) {
    const int c = blockIdx.x * 256 + threadIdx.x;
    float s = 0.f, s2 = 0.f;
    for (int k = 0; k < K; ++k) {
        float v = nodes[(size_t)k * C + c];
        s += v;
        s2 = fmaf(v, v, s2);
    }
    float mean = s / (float)K;
    float var  = s2 / (float)K - mean * mean;
    float rstd = rsqrtf(var + EPS);
    float sc = g2[c] * rstd, sh = b2[c] - mean * sc;
    for (int k = 0; k < K; ++k)
        ndtmp[(size_t)k * C + c] = nodes[(size_t)k * C + c] * sc + sh;
}

// ---------------- Stage 2b: linear_1 + bn_3 fused (one block per node row) ----------------
__global__ __launch_bounds__(256) void k_lin3(const float* __restrict__ ndtmp,
                                              const float* __restrict__ W,
                                              const float* __restrict__ bW,
                                              const float* __restrict__ g3,
                                              const float* __restrict__ b3,
                                              float* __restrict__ ndh,
                                              float* __restrict__ ndhT) {
    __shared__ float row[C];
    __shared__ float sm[256];
    const int k = blockIdx.x, tid = threadIdx.x;
    for (int i = tid; i < C; i += 256) row[i] = ndtmp[(size_t)k * C + i];
    __syncthreads();
    float out[4];
    for (int j4 = 0; j4 < 4; ++j4) {
        int j = tid + j4 * 256;
        const float* wr = W + (size_t)j * C;  // W.T matmul => dot(row, W[j,:])
        float acc = bW[j];
        for (int c = 0; c < C; ++c) acc = fmaf(row[c], wr[c], acc);
        out[j4] = acc;
    }
    float s  = out[0] + out[1] + out[2] + out[3];
    float s2 = out[0]*out[0] + out[1]*out[1] + out[2]*out[2] + out[3]*out[3];
    s  = block_reduce(s,  sm);
    s2 = block_reduce(s2, sm);
    float mean = s / (float)C;
    float var  = s2 / (float)C - mean * mean;
    float rstd = rsqrtf(var + EPS);
    float sc = g3[k] * rstd, sh = b3[k] - mean * sc;
    for (int j4 = 0; j4 < 4; ++j4) {
        int j = tid + j4 * 256;
        float v = out[j4] * sc + sh;
        ndh[(size_t)k * C + j]  = v;
        ndhT[(size_t)j * K + k] = v;
    }
}

// ---------------- Stage 3: alpha[n] = gate * (ndh @ xhat[n])  (WMMA f32 16x16x4) ----------------
// Block: 256 thr = 8 waves; covers full M=128 (nodes) x 32 cols of S. Grid: (49, 32).
// LDS holds B in K-pair-major v2f layout so each lane's B fragment is ONE ds_load_b64.
__global__ __launch_bounds__(256) void k_gemm1(const float* __restrict__ x,
                                               const float* __restrict__ sc1,
                                               const float* __restrict__ sh1,
                                               const float* __restrict__ ndh,
                                               const float* __restrict__ loc,
                                               float* __restrict__ alpha) {
    __shared__ v2f Bsh[32][33];            // [kpair (64 K-rows / 2)][32 cols + pad]
    const int n   = blockIdx.y;
    const int s0  = blockIdx.x * 32;
    const int tid = threadIdx.x;
    const int lane = tid & 31;
    const int wave = tid >> 5;
    const int col  = lane & 15;            // B/D column, also A row within tile
    const int halfsel = lane >> 4;         // 0 or 1: selects K sub-pair per ISA layout
    const int kq   = halfsel * 2;
    const int mrow = wave * 16 + col;      // node index (GEMM M)
    v8f acc0 = {}; v8f acc1 = {};
    for (int c0 = 0; c0 < C; c0 += 64) {
        // stage xhat chunk (bn1 applied on load); pair two consecutive K-rows per v2f
        for (int i = 0; i < 4; ++i) {
            int idx = tid + i * 256;
            int kp = idx >> 5, cc = idx & 31;
            int c = c0 + kp * 2;
            const float* px = x + (size_t)(n * C + c) * S + s0 + cc;
            v2f v;
            v[0] = px[0] * sc1[c]     + sh1[c];
            v[1] = px[S] * sc1[c + 1] + sh1[c + 1];
            Bsh[kp][cc] = v;                           // one ds_store_b64
        }
        __syncthreads();
        const float* arow = ndh + (size_t)mrow * C + c0;
        for (int kk = 0; kk < 64; kk += 4) {
            v2f a  = *(const v2f*)(arow + kk + kq);    // global_load_b64 (L2-hot)
            int kp = (kk >> 1) + halfsel;
            v2f b0 = Bsh[kp][col];                     // one ds_load_b64
            v2f b1 = Bsh[kp][col + 16];                // one ds_load_b64
            acc0 = __builtin_amdgcn_wmma_f32_16x16x4_f32(false, a, false, b0, (short)0, acc0, false, false);
            acc1 = __builtin_amdgcn_wmma_f32_16x16x4_f32(false, a, false, b1, (short)0, acc1, false, false);
        }
        __syncthreads();
    }
    // deterministic hard-concrete gate
    float sb = 1.0f / (1.0f + __expf(-loc[0]));
    float gate = sb * 1.2f - 0.1f;
    gate = fminf(fmaxf(gate, 0.f), 1.f);
    for (int v = 0; v < 8; ++v) {
        int m = wave * 16 + v + halfsel * 8;           // C/D row per ISA layout
        alpha[(size_t)(n * K + m) * S + s0 + col]      = acc0[v] * gate;
        alpha[(size_t)(n * K + m) * S + s0 + 16 + col] = acc1[v] * gate;
    }
}

// ---------------- Stage 4: alpha sufficient statistics: v[k], M[k1,k2] ----------------
__global__ __launch_bounds__(256) void k_astats(const float* __restrict__ alpha,
                                                float* __restrict__ vsum,
                                                float* __restrict__ Msum) {
    const int bi = blockIdx.y, bj = blockIdx.x;
    const int ti = threadIdx.x >> 4, tj = threadIdx.x & 15;
    const int k1 = bi * 16 + ti, k2 = bj * 16 + tj;
    float m = 0.f, vs = 0.f;
    for (int n = 0; n < Nn; ++n) {
        const float* p1 = alpha + (size_t)(n * K + k1) * S;
        const float* p2 = alpha + (size_t)(n * K + k2) * S;
        for (int s = 0; s < S; ++s) {
            float a1 = p1[s];
            m  = fmaf(a1, p2[s], m);
            vs += a1;
        }
    }
    Msum[k1 * K + k2] = m;
    if (bj == 0 && tj == 0) vsum[k1] = vs;
}

// ---------------- Stage 5: fold stats into bn_6 scale/shift (no pass over y!) ----------------
__global__ void k_bn6(const float* __restrict__ ndhT,
                      const float* __restrict__ vsum,
                      const float* __restrict__ Msum,
                      const float* __restrict__ g6,
                      const float* __restrict__ b6,
                      float* __restrict__ sc6,
                      float* __restrict__ sh6) {
    const int c = blockIdx.x * 128 + threadIdx.x;
    const float* ncol = ndhT + (size_t)c * K;
    float mv = 0.f;
    for (int k = 0; k < K; ++k) mv = fmaf(ncol[k], vsum[k], mv);
    float meanY = mv / (float)NS;
    float e2 = 0.f;
    for (int k1 = 0; k1 < K; ++k1) {
        const float* mrow = Msum + k1 * K;
        float t = 0.f;
        for (int k2 = 0; k2 < K; ++k2) t = fmaf(mrow[k2], ncol[k2], t);
        e2 = fmaf(ncol[k1], t, e2);
    }
    e2 /= (float)NS;
    float var  = e2 - meanY * meanY;
    float rstd = rsqrtf(var + EPS);
    float sc = g6[c] * rstd;
    sc6[c] = sc;
    sh6[c] = b6[c] - meanY * sc;
}

// ---------------- Stage 6: y[n] = ndh^T @ alpha[n], bn_6 + LeakyReLU epilogue ----------------
// Block: 8 waves cover 128 channels x 32 cols. Grid: (49, 8, 32).
__global__ __launch_bounds__(256) void k_gemm2(const float* __restrict__ alpha,
                                               const float* __restrict__ ndhT,
                                               const float* __restrict__ sc6,
                                               const float* __restrict__ sh6,
                                               float* __restrict__ y) {
    __shared__ v2f Bsh[64][33];            // [kpair (128 K-rows / 2)][32 cols + pad]
    const int n    = blockIdx.z;
    const int cblk = blockIdx.y * 128;
    const int s0   = blockIdx.x * 32;
    const int tid  = threadIdx.x;
    const int lane = tid & 31;
    const int wave = tid >> 5;
    const int col  = lane & 15;
    const int halfsel = lane >> 4;
    const int kq   = halfsel * 2;
    // stage full K x 32 chunk of alpha, paired layout
    for (int i = 0; i < 8; ++i) {
        int idx = tid + i * 256;
        int kp = idx >> 5, cc = idx & 31;
        const float* pa = alpha + (size_t)(n * K + kp * 2) * S + s0 + cc;
        v2f v;
        v[0] = pa[0];
        v[1] = pa[S];
        Bsh[kp][cc] = v;                               // one ds_store_b64
    }
    __syncthreads();
    const int crow = cblk + wave * 16 + col;           // channel index (GEMM M)
    const float* arow = ndhT + (size_t)crow * K;
    v8f acc0 = {}; v8f acc1 = {};
    for (int kk = 0; kk < K; kk += 4) {
        v2f a  = *(const v2f*)(arow + kk + kq);
        int kp = (kk >> 1) + halfsel;
        v2f b0 = Bsh[kp][col];
        v2f b1 = Bsh[kp][col + 16];
        acc0 = __builtin_amdgcn_wmma_f32_16x16x4_f32(false, a, false, b0, (short)0, acc0, false, false);
        acc1 = __builtin_amdgcn_wmma_f32_16x16x4_f32(false, a, false, b1, (short)0, acc1, false, false);
    }
    for (int v = 0; v < 8; ++v) {
        int cch = cblk + wave * 16 + v + halfsel * 8;
        float sc = sc6[cch], sh = sh6[cch];
        float y0 = acc0[v] * sc + sh;
        float y1 = acc1[v] * sc + sh;
        y0 = (y0 >= 0.f) ? y0 : 0.2f * y0;
        y1 = (y1 >= 0.f) ? y1 : 0.2f * y1;
        y[(size_t)(n * C + cch) * S + s0 + col]      = y0;
        y[(size_t)(n * C + cch) * S + s0 + 16 + col] = y1;
    }
}

extern "C" void kernel_launch(void* const* d_in, const int* in_sizes, int n_in,
                              void* d_out, int out_size, void* d_ws, size_t ws_size,
                              hipStream_t stream) {
    const float* x     = (const float*)d_in[0];
    const float* nodes = (const float*)d_in[1];
    const float* g1    = (const float*)d_in[2];
    const float* b1    = (const float*)d_in[3];
    const float* g2    = (const float*)d_in[4];
    const float* b2    = (const float*)d_in[5];
    const float* g3    = (const float*)d_in[6];
    const float* b3    = (const float*)d_in[7];
    const float* W     = (const float*)d_in[8];
    const float* bW    = (const float*)d_in[9];
    const float* loc   = (const float*)d_in[10];
    const float* g6    = (const float*)d_in[11];
    const float* b6    = (const float*)d_in[12];
    float* out = (float*)d_out;

    float* ws    = (float*)d_ws;
    float* sc1   = ws;                  // 1024
    float* sh1   = sc1 + 1024;          // 1024
    float* sc6   = sh1 + 1024;          // 1024
    float* sh6   = sc6 + 1024;          // 1024
    float* ndtmp = sh6 + 1024;          // 128*1024
    float* ndh   = ndtmp + K * C;       // 128*1024
    float* ndhT  = ndh + K * C;         // 1024*128
    float* vsum  = ndhT + C * K;        // 128
    float* Msum  = vsum + K;            // 128*128
    float* alpha = Msum + K * K;        // 32*128*1568 (~25.7 MB; total ws ~27.4 MB)

    k_bn1  <<<C, 256, 0, stream>>>(x, g1, b1, sc1, sh1);
    k_bn2  <<<C / 256, 256, 0, stream>>>(nodes, g2, b2, ndtmp);
    k_lin3 <<<K, 256, 0, stream>>>(ndtmp, W, bW, g3, b3, ndh, ndhT);
    k_gemm1<<<dim3(S / 32, Nn), 256, 0, stream>>>(x, sc1, sh1, ndh, loc, alpha);
    k_astats<<<dim3(K / 16, K / 16), 256, 0, stream>>>(alpha, vsum, Msum);
    k_bn6  <<<C / 128, 128, 0, stream>>>(ndhT, vsum, Msum, g6, b6, sc6, sh6);
    k_gemm2<<<dim3(S / 32, C / 128, Nn), 256, 0, stream>>>(alpha, ndhT, sc6, sh6, out);
}